// Embedding_2233382994081
// MI455X (gfx1250) — compile-verified
//
#include <hip/hip_runtime.h>
#include <math.h>

// CDNA5 wave32 WMMA vector types
typedef __attribute__((ext_vector_type(2))) float v2f;   // A/B of V_WMMA_F32_16X16X4_F32 (64 f32 / 32 lanes)
typedef __attribute__((ext_vector_type(8))) float v8f;   // 16x16 f32 C/D (256 f32 / 32 lanes)

// Problem constants (from reference): B=8, N=2048 (= 4 heads x q=512), L=2048, d=512, top-k=3
#define LQ   512
#define LL   2048
#define DD   512

__global__ __launch_bounds__(256)
void embed_topk_wmma_kernel(const float* __restrict__ logits,
                            const float* __restrict__ emb0,
                            const float* __restrict__ emb1,
                            const float* __restrict__ emb2,
                            const float* __restrict__ emb3,
                            float* __restrict__ out)
{
  // Per-block: one b, 64 consecutive l.
  __shared__ float wbuf[64][4][3];   // softmax weights  [l][head][k]
  __shared__ int   ibuf[64][4][3];   // top-3 indices    [l][head][k]
  __shared__ float stg[8][16][20];   // per-wave 16x16 transpose patch, stride 20 => bank-conflict free

  const int b  = blockIdx.y;
  const int l0 = blockIdx.x << 6;
  const int t  = threadIdx.x;

  // ---------------- Phase 1: streaming top-3 + softmax (HBM-bound part 1) ----------------
  {
    const int l = t & 63;            // lanes -> consecutive l => 128B coalesced loads per wave
    const int h = t >> 6;            // head 0..3
    const float* p = logits + ((size_t)(b * 2048 + h * LQ)) * LL + (size_t)(l0 + l);
    float v1 = -INFINITY, v2 = -INFINITY, v3 = -INFINITY;
    int   i1 = 0, i2 = 0, i3 = 0;
    #pragma unroll 8
    for (int j = 0; j < LQ; ++j) {
      // 8KB-stride stream: HW prefetcher can't follow it; speculative global_prefetch is safe past end
      __builtin_prefetch(p + (size_t)(j + 16) * LL, 0, 0);
      const float v = p[j * LL];     // 32-bit offset: base VGPR pair + immediate IOFFSET per unrolled step
      if (v > v1)      { v3 = v2; i3 = i2; v2 = v1; i2 = i1; v1 = v; i1 = j; }
      else if (v > v2) { v3 = v2; i3 = i2; v2 = v;  i2 = j; }
      else if (v > v3) { v3 = v;  i3 = j; }
    }
    // softmax over the (descending) top-3
    const float e2 = expf(v2 - v1);
    const float e3 = expf(v3 - v1);
    const float r  = 1.0f / (1.0f + e2 + e3);
    wbuf[l][h][0] = r;
    wbuf[l][h][1] = e2 * r;
    wbuf[l][h][2] = e3 * r;
    ibuf[l][h][0] = i1; ibuf[l][h][1] = i2; ibuf[l][h][2] = i3;
  }
  __syncthreads();

  // ---------------- Phase 2: exact-f32 sparse combine on the matrix pipe ----------------
  // Union-basis trick: per (head, 16-l group), the 48 gathered rows are a K=48 basis.
  // A[l, j] = w[l, j%3] iff j/3 == l else 0 ;  B[j, d] = emb_h[idx[j/3, j%3], d]
  // => D = A x B reproduces  sum_k w[l,k]*emb[idx[l,k], :]  exactly, via 12 K=4 f32 WMMAs.
  const int wave = t >> 5;
  const int lane = t & 31;
  const int lg   = wave & 3;         // which 16-l group of the 64
  const int dh   = wave >> 2;        // d half: waves 0..3 -> d[0,256), waves 4..7 -> d[256,512)
  const int m    = lane & 15;        // A row / B,D column (ISA 7.12.2 layouts)
  const int kh   = lane >> 4;        // lane half: K = 2*kh + vgpr for A and B
  const int lg16 = lg << 4;

  const float* embs[4] = { emb0, emb1, emb2, emb3 };
  float* const outb = out + (size_t)b * DD * LL;

  for (int dtc = 0; dtc < 4; ++dtc) {          // 4 chunks of 4 d-tiles each
    const int dbase = (dh * 16 + dtc * 4) * 16;
    v8f acc[4];
    #pragma unroll
    for (int tl = 0; tl < 4; ++tl) acc[tl] = (v8f){0,0,0,0,0,0,0,0};

    for (int s = 0; s < 12; ++s) {             // 12 K=4 steps cover the 48-row basis
      const int j0 = 4 * s + 2 * kh;           // basis row for this lane's K = 2*kh + 0
      const int j1 = j0 + 1;                   // and K = 2*kh + 1
      const int o0 = j0 / 3, k0 = j0 - 3 * o0; // owning l-row, which of its 3 weights (amortized over 4 heads)
      const int o1 = j1 / 3, k1 = j1 - 3 * o1;
      const bool own0 = (o0 == m);
      const bool own1 = (o1 == m);
      #pragma unroll
      for (int h = 0; h < 4; ++h) {
        // Unconditional LDS loads (addresses always valid) + v_cndmask select: no EXEC games
        const float w0 = wbuf[lg16 + m][h][k0];
        const float w1 = wbuf[lg16 + m][h][k1];
        v2f a;
        a.x = own0 ? w0 : 0.0f;
        a.y = own1 ? w1 : 0.0f;
        const int idx0 = ibuf[lg16 + o0][h][k0];
        const int idx1 = ibuf[lg16 + o1][h][k1];
        const float* e = embs[h];
        // 32-bit element offsets from the uniform emb base => saddr+voffset global loads
        const int off0 = idx0 * DD + (dbase + m);
        const int off1 = idx1 * DD + (dbase + m);
        v2f bb[4];
        #pragma unroll
        for (int tl = 0; tl < 4; ++tl) {       // 8 independent loads -> one clause, one wait
          bb[tl].x = e[off0 + tl * 16];        // 64B-coalesced per half-wave, L2-resident (4MB codebooks)
          bb[tl].y = e[off1 + tl * 16];
        }
        #pragma unroll
        for (int tl = 0; tl < 4; ++tl)
          acc[tl] = __builtin_amdgcn_wmma_f32_16x16x4_f32(
              false, a, false, bb[tl], (short)0, acc[tl], false, false);
      }
    }

    // Stage each 16x16 D tile through LDS and emit (B, d, L)-transposed, coalesced stores.
    // Same-wave producer/consumer: DS ops are in-order per wave, no barrier needed.
    #pragma unroll
    for (int tl = 0; tl < 4; ++tl) {
      const int d0 = dbase + tl * 16;
      #pragma unroll
      for (int v = 0; v < 8; ++v)
        stg[wave][v + 8 * kh][m] = acc[tl][v];       // (l_local, d_local) = (v+8*kh, m)
      #pragma unroll
      for (int dd = 0; dd < 8; ++dd) {
        const int dl = dd * 2 + kh;                  // local d row
        outb[(size_t)(d0 + dl) * LL + (size_t)(l0 + lg16 + m)] = stg[wave][m][dl];
      }
    }
  }
}

extern "C" void kernel_launch(void* const* d_in, const int* in_sizes, int n_in,
                              void* d_out, int out_size, void* d_ws, size_t ws_size,
                              hipStream_t stream) {
  const float* logits = (const float*)d_in[0];
  const float* e0 = (const float*)d_in[1];
  const float* e1 = (const float*)d_in[2];
  const float* e2 = (const float*)d_in[3];
  const float* e3 = (const float*)d_in[4];
  float* outp = (float*)d_out;

  dim3 grid(LL / 64, 8, 1);   // 32 l-tiles x B=8 -> 256 blocks of 256 threads (8 wave32s)
  embed_topk_wmma_kernel<<<grid, 256, 0, stream>>>(logits, e0, e1, e2, e3, outp);

  (void)in_sizes; (void)n_in; (void)out_size; (void)d_ws; (void)ws_size;
}